// ProteinCRF_91190745628897
// MI455X (gfx1250) — compile-verified
//
#include <hip/hip_runtime.h>
#include <math.h>

// Problem constants (from reference): B=2048, L=2048, T=8
#define Bsz   2048
#define Lsz   2048
#define Tt    8
#define CHUNK 128
#define NC    16      // ceil((L-1)/CHUNK) = 2047/128 -> 16 chunks over steps t=1..L-1

typedef __attribute__((ext_vector_type(2))) float v2f;
typedef __attribute__((ext_vector_type(8))) float v8f;

// ---------------------------------------------------------------------------
// Kernel 1: per-(row,chunk) log-semiring 8x8 matrix product, two rows per wave.
//
// State R (8x8 per task, two tasks per wave) is kept in linear space with a
// power-of-2 scale counter sExp.  One step:
//   R' = S_t * R,   S_t[j,k] = exp(em[b,t,j]) * exp(trans[k][j])
// realized as D(16x16) = A(16x8) x B(8x16) via two chained
// v_wmma_f32_16x16x4_f32 (K=0..3 then K=4..7), with the two tasks packed
// block-diagonally: task0 -> D[0:8,0:8], task1 -> D[8:16,8:16].
//
// Lane conventions (wave32):  h = lane>>4, p = lane&15, task = p>>3, sub = p&7
//   A operand : VGPR r, half h -> K = kbase + 2h + r ; lane holds A[M=p][K]
//   B operand : VGPR q, half h -> K = ((q&2)<<1) + 2h + (q&1) ; lane holds B[K][N=p]
//   C/D       : VGPR v, half h -> M = v + 8h ; lane holds D[M][N=p]
// D -> next-B relayout is a fixed lane permutation: srcl = p + ((p&8)<<1).
//
// Loop bounds are forced wave-uniform (readfirstlane) so loop control is pure
// SALU and EXEC is provably all-ones at every WMMA issue (ISA requirement).
// ---------------------------------------------------------------------------
__global__ __launch_bounds__(256)
void crf_chunk_kernel(const float* __restrict__ em,
                      const float* __restrict__ trans,
                      const int*   __restrict__ tags,
                      float* __restrict__ wsM,
                      float* __restrict__ wsNum,
                      int*   __restrict__ wsCnt,
                      int*   __restrict__ wsLast)
{
    __shared__ float ldsT[Tt * Tt];
    const int tid = threadIdx.x;
    if (tid < Tt * Tt) ldsT[tid] = trans[tid];
    __syncthreads();

    const int lane = tid & 31;
    const int gw   = blockIdx.x * 8 + (tid >> 5);   // global wave id: 0..16383
    const int c    = gw & (NC - 1);                 // chunk index
    const int pair = gw >> 4;                       // row pair index
    const int b0   = pair * 2;

    const int h    = lane >> 4;
    const int p    = lane & 15;
    const int task = p >> 3;
    const int sub  = p & 7;
    const int bl   = b0 + task;                     // this lane's batch row

    // Constant part of A: cA[2w+r] = exp(trans[k][sub]), k = 4w + 2h + r
    float cA[4];
#pragma unroll
    for (int w = 0; w < 2; ++w)
#pragma unroll
        for (int r = 0; r < 2; ++r)
            cA[w * 2 + r] = __expf(trans[(4 * w + 2 * h + r) * Tt + sub]);

    // Canonical state in B-operand layout, initialized to identity.
    float bb[4];
#pragma unroll
    for (int q = 0; q < 4; ++q) {
        const int k = ((q & 2) << 1) + 2 * h + (q & 1);
        bb[q] = (k == sub) ? 1.0f : 0.0f;
    }
    int sExp = 0;

    // Wave-uniform (all lanes of a wave share c): force into SGPRs so the
    // main loop is SALU-controlled and EXEC stays all-ones for WMMA.
    const int t0 = __builtin_amdgcn_readfirstlane(1 + c * CHUNK);
    const int t1 = __builtin_amdgcn_readfirstlane(
                       (t0 + CHUNK < Lsz) ? (t0 + CHUNK) : Lsz);

    const int srcl = p + ((p & 8) << 1);            // D->B gather source lane

    int   prevTg  = tags[(size_t)bl * Lsz + (t0 - 1)];
    float accNum  = 0.0f;
    int   cnt     = 0;
    int   lastTag = -1;

#pragma unroll 2
    for (int t = t0; t < t1; ++t) {
        const int   tg  = tags[(size_t)bl * Lsz + t];
        const float emv = em[((size_t)bl * Lsz + t) * Tt + sub];
        const bool  msk = (tg != 0);
        const float e   = __expf(emv);

        // Build A (row scaling by exp(em) is per-lane since lane == M)
        v2f a1, a2, bo1, bo2;
        a1.x = cA[0] * e;  a1.y = cA[1] * e;
        a2.x = cA[2] * e;  a2.y = cA[3] * e;
        bo1.x = bb[0];     bo1.y = bb[1];
        bo2.x = bb[2];     bo2.y = bb[3];

        v8f d = {};
        d = __builtin_amdgcn_wmma_f32_16x16x4_f32(false, a1, false, bo1,
                                                  (short)0, d, false, false);
        d = __builtin_amdgcn_wmma_f32_16x16x4_f32(false, a2, false, bo2,
                                                  (short)0, d, false, false);

        // Relayout D -> B-operand layout for the next step (+ per-task mask select)
        float nb[4];
#pragma unroll
        for (int q = 0; q < 4; ++q) {
            const int klo = ((q & 2) << 1) + (q & 1);       // row for lanes < 16
            const float lo = __shfl(d[klo],     srcl, 32);
            const float hi = __shfl(d[klo + 2], srcl, 32);
            nb[q] = h ? hi : lo;
        }
#pragma unroll
        for (int q = 0; q < 4; ++q) bb[q] = msk ? nb[q] : bb[q];

        // Numerator partial (count each (row,step) once: lanes of half 0)
        const int pi = (prevTg > 0 ? prevTg : 1) - 1;
        const int ci = (tg     > 0 ? tg     : 1) - 1;
        const float tv = ldsT[pi * Tt + ci];
        if ((h == 0) && msk && (sub == ci)) accNum += emv + tv;

        cnt    += msk ? 1 : 0;
        lastTag = msk ? (tg - 1) : lastTag;
        prevTg  = msk ? tg : prevTg;

        // Power-of-2 renormalization every 8 steps (no transcendentals).
        // (t - t0) is SALU-known -> scalar branch, taken 1/8 of iterations.
        if (((t - t0) & 7) == 7) {
            float mx = fmaxf(fmaxf(bb[0], bb[1]), fmaxf(bb[2], bb[3]));
            mx = fmaxf(mx, __shfl_xor(mx, 1, 32));
            mx = fmaxf(mx, __shfl_xor(mx, 2, 32));
            mx = fmaxf(mx, __shfl_xor(mx, 4, 32));
            mx = fmaxf(mx, __shfl_xor(mx, 16, 32));  // stay inside the task's lanes
            if (mx > 0.0f) {
                int ex;
                (void)frexpf(mx, &ex);
                const float s = ldexpf(1.0f, -ex);
#pragma unroll
                for (int q = 0; q < 4; ++q) bb[q] *= s;
                sExp += ex;
            }
        }
    }

    // Emit log-space chunk matrix: logR[k][i] = log(bb) + sExp*ln2
    const float lsc = (float)sExp * 0.69314718055994531f;
#pragma unroll
    for (int q = 0; q < 4; ++q) {
        const int k = ((q & 2) << 1) + 2 * h + (q & 1);
        wsM[(((size_t)bl * NC + c) << 6) + k * Tt + sub] = __logf(bb[q]) + lsc;
    }

    // Reduce numerator partial within each task's 8-lane group (half 0)
    accNum += __shfl_xor(accNum, 1, 32);
    accNum += __shfl_xor(accNum, 2, 32);
    accNum += __shfl_xor(accNum, 4, 32);
    if (lane == 0 || lane == 8) {
        const int idx = bl * NC + c;
        wsNum[idx]  = accNum;
        wsCnt[idx]  = cnt;
        wsLast[idx] = lastTag;
    }
}

// ---------------------------------------------------------------------------
// Kernel 2: per-row composition of the 16 chunk matrices with alpha0,
// numerator assembly, per-row log-likelihood.
// Stored matrix is logR[j][i]: newA[j] = logsumexp_i(alpha[i] + logR[j][i]).
// ---------------------------------------------------------------------------
__global__ __launch_bounds__(256)
void crf_combine_kernel(const float* __restrict__ em,
                        const float* __restrict__ startT,
                        const float* __restrict__ endT,
                        const int*   __restrict__ tags,
                        const float* __restrict__ wsM,
                        const float* __restrict__ wsNum,
                        const int*   __restrict__ wsCnt,
                        const int*   __restrict__ wsLast,
                        float* __restrict__ wsRow)
{
    const int b = blockIdx.x * blockDim.x + threadIdx.x;
    if (b >= Bsz) return;

    float alpha[Tt];
#pragma unroll
    for (int j = 0; j < Tt; ++j)
        alpha[j] = startT[j] + em[(size_t)b * Lsz * Tt + j];

    const int tg0 = tags[(size_t)b * Lsz];          // always >= 1 (lengths >= L/2)
    const int s0  = tg0 - 1;
    float num     = startT[s0] + em[(size_t)b * Lsz * Tt + s0];
    int   lastTag = s0;

    for (int cI = 0; cI < NC; ++cI) {
        const float* M = wsM + (((size_t)b * NC + cI) << 6);
        float na[Tt];
#pragma unroll
        for (int j = 0; j < Tt; ++j) {
            float r[Tt];
#pragma unroll
            for (int i = 0; i < Tt; ++i) r[i] = alpha[i] + M[j * Tt + i];
            float mx = r[0];
#pragma unroll
            for (int i = 1; i < Tt; ++i) mx = fmaxf(mx, r[i]);
            float s = 0.0f;
#pragma unroll
            for (int i = 0; i < Tt; ++i) s += __expf(r[i] - mx);
            na[j] = mx + __logf(s);
        }
#pragma unroll
        for (int j = 0; j < Tt; ++j) alpha[j] = na[j];

        num += wsNum[b * NC + cI];
        if (wsCnt[b * NC + cI] > 0) lastTag = wsLast[b * NC + cI];
    }

    num += endT[lastTag];

    float r[Tt];
#pragma unroll
    for (int j = 0; j < Tt; ++j) r[j] = alpha[j] + endT[j];
    float mx = r[0];
#pragma unroll
    for (int j = 1; j < Tt; ++j) mx = fmaxf(mx, r[j]);
    float s = 0.0f;
#pragma unroll
    for (int j = 0; j < Tt; ++j) s += __expf(r[j] - mx);
    const float denom = mx + __logf(s);

    wsRow[b] = num - denom;                          // llh for row b
}

// ---------------------------------------------------------------------------
// Kernel 3: mean-reduce and negate.
// ---------------------------------------------------------------------------
__global__ __launch_bounds__(256)
void crf_reduce_kernel(const float* __restrict__ wsRow, float* __restrict__ out)
{
    __shared__ float sm[256];
    const int tid = threadIdx.x;
    float s = 0.0f;
    for (int i = tid; i < Bsz; i += 256) s += wsRow[i];
    sm[tid] = s;
    __syncthreads();
    for (int st = 128; st > 0; st >>= 1) {
        if (tid < st) sm[tid] += sm[tid + st];
        __syncthreads();
    }
    if (tid == 0) out[0] = -sm[0] / (float)Bsz;
}

// ---------------------------------------------------------------------------
// Host-side launcher.  Inputs (setup_inputs order):
//   0: emissions (B,L,T) f32   1: transitions (T,T) f32
//   2: start_transitions (T)   3: end_transitions (T)   4: tags (B,L) int32
// ---------------------------------------------------------------------------
extern "C" void kernel_launch(void* const* d_in, const int* in_sizes, int n_in,
                              void* d_out, int out_size, void* d_ws, size_t ws_size,
                              hipStream_t stream)
{
    (void)in_sizes; (void)n_in; (void)out_size; (void)ws_size;

    const float* em     = (const float*)d_in[0];
    const float* trans  = (const float*)d_in[1];
    const float* startT = (const float*)d_in[2];
    const float* endT   = (const float*)d_in[3];
    const int*   tags   = (const int*)  d_in[4];
    float* out = (float*)d_out;

    // Workspace carve-up (all 4-byte aligned; ~8.8 MB total)
    char* ws = (char*)d_ws;
    float* wsM    = (float*)ws;                                   // B*NC*64 f32
    size_t off    = (size_t)Bsz * NC * 64 * sizeof(float);
    float* wsNum  = (float*)(ws + off);  off += (size_t)Bsz * NC * sizeof(float);
    int*   wsCnt  = (int*)  (ws + off);  off += (size_t)Bsz * NC * sizeof(int);
    int*   wsLast = (int*)  (ws + off);  off += (size_t)Bsz * NC * sizeof(int);
    float* wsRow  = (float*)(ws + off);

    // 1) 32768 chunk tasks, 2 per wave, 8 waves per 256-thread block
    const int waves  = (Bsz / 2) * NC;        // 16384
    const int blocks = waves / 8;             // 2048
    crf_chunk_kernel<<<blocks, 256, 0, stream>>>(em, trans, tags,
                                                 wsM, wsNum, wsCnt, wsLast);

    // 2) one thread per batch row
    crf_combine_kernel<<<(Bsz + 255) / 256, 256, 0, stream>>>(
        em, startT, endT, tags, wsM, wsNum, wsCnt, wsLast, wsRow);

    // 3) final scalar
    crf_reduce_kernel<<<1, 256, 0, stream>>>(wsRow, out);
}